// DynamicLinkPredictor_21680994910620
// MI455X (gfx1250) — compile-verified
//
#include <hip/hip_runtime.h>
#include <math.h>

// Problem constants (match reference)
#define BB 2
#define NN 512
#define FF 6
#define TT 10
#define HH 64
#define HD2 32
#define EE 16384          // 2*E0 symmetrized edges
#define BN (BB*NN)        // 1024 rows
#define BNH (BN*HH)       // 65536
#define NPAIR (BB*NN*NN)  // 524288
#define EPSLN 1e-5f
#define PW 4              // waves per block in pair kernel

// staged (transposed) weight offsets in LDS for the pair kernel
#define OFF_R1WS 0        // 32 x 64
#define OFF_R1W1 2048     // 32 x 64
#define OFF_R1W2 4096     // 32 x 32
#define OFF_R2W1 5120     // 32 x 32
#define OFF_R2W2 6144     // 32 x 32
#define SW_TOTAL 7168

typedef float v2f __attribute__((ext_vector_type(2)));
typedef float v8f __attribute__((ext_vector_type(8)));

#if defined(__gfx1250__) && __has_builtin(__builtin_amdgcn_global_load_async_to_lds_b32) && \
    __has_builtin(__builtin_amdgcn_s_wait_asynccnt)
#define ASYNC_LDS 1
#else
#define ASYNC_LDS 0
#endif

#if ASYNC_LDS
typedef __attribute__((address_space(1))) int* gint_p;
typedef __attribute__((address_space(3))) int* lint_p;
__device__ __forceinline__ void async_copy_f32(const float* g, float* l) {
  __builtin_amdgcn_global_load_async_to_lds_b32((gint_p)g, (lint_p)l, 0, 0);
}
#endif

__device__ __forceinline__ v8f wmma4(v8f c, v2f a, v2f b) {
  // D = A(16x4,f32) x B(4x16,f32) + C(16x16,f32)
  return __builtin_amdgcn_wmma_f32_16x16x4_f32(false, a, false, b, (short)0, c, false, false);
}
__device__ __forceinline__ float sigmoidf_(float x) { return 1.f / (1.f + expf(-x)); }

// ---------------------------------------------------------------- utility
__global__ void k_zero(float* __restrict__ p, int n) {
  int i = blockIdx.x * blockDim.x + threadIdx.x;
  if (i < n) p[i] = 0.f;
}
__global__ void k_negcopy(float* __restrict__ d, const float* __restrict__ s, int n) {
  int i = blockIdx.x * blockDim.x + threadIdx.x;
  if (i < n) d[i] = -s[i];
}

// ---------------------------------------------------------------- graph prep
__global__ void k_deg(const float* __restrict__ ew, const int* __restrict__ dst,
                      float* __restrict__ deg) {
  int e = blockIdx.x * blockDim.x + threadIdx.x;
  if (e < EE) atomicAdd(&deg[dst[e]], ew[e]);
}
__global__ void k_nodenorm(const float* __restrict__ deg, float* __restrict__ dinv,
                           float* __restrict__ dgi, float* __restrict__ gself) {
  int i = blockIdx.x * blockDim.x + threadIdx.x;
  if (i < NN) {
    float d = deg[i];
    dinv[i] = (d > 0.f) ? rsqrtf(fmaxf(d, 1e-12f)) : 0.f;
    float q = rsqrtf(d + 2.f);
    dgi[i] = q;
    gself[i] = 2.f * q * q;
  }
}
__global__ void k_edgew(const float* __restrict__ ew, const int* __restrict__ src,
                        const int* __restrict__ dst, const float* __restrict__ dinv,
                        const float* __restrict__ dgi, float* __restrict__ cw,
                        float* __restrict__ gw) {
  int e = blockIdx.x * blockDim.x + threadIdx.x;
  if (e < EE) {
    int s = src[e], d = dst[e];
    float w = ew[e];
    cw[e] = -dinv[s] * w * dinv[d];
    gw[e] = dgi[s] * w * dgi[d];
  }
}

// P[b,dst,c] += mult * w[e] * V[b,src,c]
__global__ void k_prop(float* __restrict__ P, const float* __restrict__ V,
                       const float* __restrict__ w, const int* __restrict__ src,
                       const int* __restrict__ dst, float mult) {
  int idx = blockIdx.x * blockDim.x + threadIdx.x; // EE*BB*HH threads
  if (idx >= EE * BB * HH) return;
  int e = idx >> 7;          // /(BB*HH)
  int r = idx & 127;
  int b = r >> 6;
  int c = r & 63;
  float v = V[(b * NN + src[e]) * HH + c];
  atomicAdd(&P[(b * NN + dst[e]) * HH + c], mult * w[e] * v);
}

// P = P + gself[n]*vp + bias[c]
__global__ void k_gcnfin(float* __restrict__ P, const float* __restrict__ vp,
                         const float* __restrict__ gself, const float* __restrict__ bias) {
  int idx = blockIdx.x * blockDim.x + threadIdx.x;
  if (idx < BNH) {
    int n = (idx >> 6) & (NN - 1);
    int c = idx & 63;
    P[idx] += gself[n] * vp[idx] + bias[c];
  }
}

// ---------------------------------------------------------------- input pack
// xp[row, t*F+f] = x[row, f, t]
__global__ void k_pack(const float* __restrict__ x, float* __restrict__ xp) {
  int idx = blockIdx.x * blockDim.x + threadIdx.x;
  if (idx >= BN * (TT * FF)) return;
  int row = idx / (TT * FF);
  int q = idx - row * (TT * FF);
  int t = q / FF, f = q - t * FF;
  xp[idx] = x[(row * FF + f) * TT + t];
}

// ---------------------------------------------------------------- WMMA GEMM
// C(Mx64) = A(MxK) @ W(Kx64) [+bias | += if acc]. grid.x=M/16, block=128 (4 waves = 4 n-tiles)
__global__ void k_gemm(const float* __restrict__ A, const float* __restrict__ W,
                       const float* __restrict__ bias, float* __restrict__ C,
                       int K, int acc) {
  const int lane = threadIdx.x & 31;
  const int ntile = threadIdx.x >> 5; // 0..3
  const int mtile = blockIdx.x;
  const int m = lane & 15;
  const int half = lane >> 4;
  const int col = ntile * 16 + m;
  const int row0 = mtile * 16;
  v8f c8;
  if (acc) {
#pragma unroll
    for (int r = 0; r < 8; ++r) c8[r] = C[(row0 + r + 8 * half) * HH + col];
  } else {
    const float bv = bias ? bias[col] : 0.f;
#pragma unroll
    for (int r = 0; r < 8; ++r) c8[r] = bv;
  }
  const float* Ar = A + (row0 + m) * K;
  for (int kk = 0; kk < K; kk += 4) {
    const int k0 = kk + 2 * half;
    v2f a; a.x = Ar[k0]; a.y = Ar[k0 + 1];
    v2f b; b.x = W[k0 * HH + col]; b.y = W[(k0 + 1) * HH + col];
    c8 = wmma4(c8, a, b);
  }
#pragma unroll
  for (int r = 0; r < 8; ++r) C[(row0 + r + 8 * half) * HH + col] = c8[r];
}

// ---------------------------------------------------------------- row LN (H=64)
// io = LN(relu? io) * g + t, per row of 64
__global__ void k_row_ln(float* __restrict__ io, const float* __restrict__ g,
                         const float* __restrict__ t, int do_relu) {
  __shared__ float red[64];
  const int row = blockIdx.x, c = threadIdx.x;
  float v = io[row * HH + c];
  if (do_relu) v = fmaxf(v, 0.f);
  red[c] = v; __syncthreads();
  for (int s = 32; s > 0; s >>= 1) { if (c < s) red[c] += red[c + s]; __syncthreads(); }
  const float mu = red[0] * (1.f / 64.f); __syncthreads();
  const float d = v - mu;
  red[c] = d * d; __syncthreads();
  for (int s = 32; s > 0; s >>= 1) { if (c < s) red[c] += red[c + s]; __syncthreads(); }
  const float inv = rsqrtf(red[0] * (1.f / 64.f) + EPSLN);
  io[row * HH + c] = d * inv * g[c] + t[c];
}

// ---------------------------------------------------------------- GRU combine
__global__ void k_gru(float* __restrict__ h, const float* __restrict__ zp,
                      const float* __restrict__ cp) {
  int i = blockIdx.x * blockDim.x + threadIdx.x;
  if (i < BNH) h[i] = (1.f - sigmoidf_(zp[i])) * tanhf(cp[i]);
}

// ---------------------------------------------------------------- LSTM combine (zero state)
__global__ void k_lstm(float* __restrict__ h, const float* __restrict__ g0,
                       const float* __restrict__ g2, const float* __restrict__ g3,
                       const float* __restrict__ bh0, const float* __restrict__ bh2,
                       const float* __restrict__ bh3, const float* __restrict__ bg0,
                       const float* __restrict__ bg2, const float* __restrict__ bg3,
                       const float* __restrict__ wc2, const float* __restrict__ lng,
                       const float* __restrict__ lnt) {
  __shared__ float red[64];
  const int row = blockIdx.x, c = threadIdx.x;
  const int idx = row * HH + c;
  const float I  = sigmoidf_(g0[idx] + bh0[c] + bg0[c]);
  const float Cn = I * tanhf(g2[idx] + bh2[c] + bg2[c]);
  const float O  = sigmoidf_(g3[idx] + bh3[c] + wc2[c] * Cn + bg3[c]);
  const float Hn = O * tanhf(Cn);
  red[c] = Hn; __syncthreads();
  for (int s = 32; s > 0; s >>= 1) { if (c < s) red[c] += red[c + s]; __syncthreads(); }
  const float mu = red[0] * (1.f / 64.f); __syncthreads();
  const float d = Hn - mu;
  red[c] = d * d; __syncthreads();
  for (int s = 32; s > 0; s >>= 1) { if (c < s) red[c] += red[c + s]; __syncthreads(); }
  const float inv = rsqrtf(red[0] * (1.f / 64.f) + EPSLN);
  h[idx] += d * inv * lng[c] + lnt[c];
}

// ---------------------------------------------------------------- fused pair MLP
struct PairParams {
  const float *lp_g, *lp_t;
  const float *r1_W1, *r1_b1, *r1_g1, *r1_t1, *r1_W2, *r1_b2, *r1_g2, *r1_t2, *r1_Ws, *r1_bs;
  const float *r2_W1, *r2_b1, *r2_g1, *r2_t1, *r2_W2, *r2_b2, *r2_g2, *r2_t2;
  const float *out_W, *out_b;
};

// two 16x16 output tiles: D(16x32) = S(16xK) @ W(Kx32) + bias.
// Wt is the TRANSPOSED weight staged in LDS: Wt[n*K + k] = W[k*32 + n]
__device__ __forceinline__ void gemm_tiles(const float* S, int ldS, const float* Wt,
                                           const float* bias, int K, int lane,
                                           v8f& t0, v8f& t1) {
  const int m = lane & 15, half = lane >> 4;
  const float b0 = bias[m], b1 = bias[16 + m];
#pragma unroll
  for (int r = 0; r < 8; ++r) { t0[r] = b0; t1[r] = b1; }
  const float* Sr = S + m * ldS;
  const float* p0 = Wt + m * K;
  const float* p1 = Wt + (16 + m) * K;
  for (int kk = 0; kk < K; kk += 4) {
    const int k0 = kk + 2 * half;
    v2f a; a.x = Sr[k0]; a.y = Sr[k0 + 1];
    v2f w; w.x = p0[k0]; w.y = p0[k0 + 1];
    t0 = wmma4(t0, a, w);
    w.x = p1[k0]; w.y = p1[k0 + 1];
    t1 = wmma4(t1, a, w);
  }
}
// same, but two weight matrices sharing one A-frag stream (resblock-1 main + skip)
__device__ __forceinline__ void gemm_tiles_dual(const float* S, int ldS,
                                                const float* Wt1, const float* bias1,
                                                const float* Wt2, const float* bias2,
                                                int K, int lane,
                                                v8f& t0, v8f& t1, v8f& s0, v8f& s1) {
  const int m = lane & 15, half = lane >> 4;
  {
    const float a0 = bias1[m], a1 = bias1[16 + m];
    const float c0 = bias2[m], c1 = bias2[16 + m];
#pragma unroll
    for (int r = 0; r < 8; ++r) { t0[r] = a0; t1[r] = a1; s0[r] = c0; s1[r] = c1; }
  }
  const float* Sr  = S + m * ldS;
  const float* p10 = Wt1 + m * K;
  const float* p11 = Wt1 + (16 + m) * K;
  const float* p20 = Wt2 + m * K;
  const float* p21 = Wt2 + (16 + m) * K;
  for (int kk = 0; kk < K; kk += 4) {
    const int k0 = kk + 2 * half;
    v2f a; a.x = Sr[k0]; a.y = Sr[k0 + 1];
    v2f w;
    w.x = p10[k0]; w.y = p10[k0 + 1]; t0 = wmma4(t0, a, w);
    w.x = p11[k0]; w.y = p11[k0 + 1]; t1 = wmma4(t1, a, w);
    w.x = p20[k0]; w.y = p20[k0 + 1]; s0 = wmma4(s0, a, w);
    w.x = p21[k0]; w.y = p21[k0 + 1]; s1 = wmma4(s1, a, w);
  }
}
__device__ __forceinline__ void store_tiles(float* D, int lane, const v8f& t0, const v8f& t1) {
  const int n = lane & 15, half = lane >> 4;
#pragma unroll
  for (int r = 0; r < 8; ++r) {
    D[(r + 8 * half) * HD2 + n]      = t0[r];
    D[(r + 8 * half) * HD2 + 16 + n] = t1[r];
  }
}
__device__ __forceinline__ void add_frag_relu(float* D, int lane, const v8f& t0, const v8f& t1) {
  const int n = lane & 15, half = lane >> 4;
#pragma unroll
  for (int r = 0; r < 8; ++r) {
    const int row = r + 8 * half;
    D[row * HD2 + n]      = fmaxf(D[row * HD2 + n] + t0[r], 0.f);
    D[row * HD2 + 16 + n] = fmaxf(D[row * HD2 + 16 + n] + t1[r], 0.f);
  }
}
__device__ __forceinline__ void ln32(float* D, const float* g, const float* t, int lane,
                                     bool do_relu) {
  if (lane < 16) {
    float* row = D + lane * HD2;
    float s = 0.f;
    for (int c = 0; c < 32; ++c) s += row[c];
    const float mu = s * (1.f / 32.f);
    float v = 0.f;
    for (int c = 0; c < 32; ++c) { float d = row[c] - mu; v += d * d; }
    const float inv = rsqrtf(v * (1.f / 32.f) + EPSLN);
    for (int c = 0; c < 32; ++c) {
      float o = (row[c] - mu) * inv * g[c] + t[c];
      row[c] = do_relu ? fmaxf(o, 0.f) : o;
    }
  }
}

__global__ void k_pairs(const float* __restrict__ hA, const float* __restrict__ hB,
                        PairParams pp, float* __restrict__ out) {
  __shared__ float sW[SW_TOTAL];     // block-wide staged transposed weights (28 KB)
  __shared__ float sZ[PW][16][64];
  __shared__ float sT[PW][16][32];
  __shared__ float sU[PW][16][32];
  __shared__ float sV[PW][16][32];
  const int wave = threadIdx.x >> 5;
  const int lane = threadIdx.x & 31;
  const int pairbase = (blockIdx.x * PW + wave) * 16;

  // stage all resblock weights (transposed) into LDS once per block
  for (int i = threadIdx.x; i < SW_TOTAL; i += blockDim.x) {
    const float* srcm; int base, K;
    if (i < OFF_R1W1)      { base = OFF_R1WS; K = 64; srcm = pp.r1_Ws; }
    else if (i < OFF_R1W2) { base = OFF_R1W1; K = 64; srcm = pp.r1_W1; }
    else if (i < OFF_R2W1) { base = OFF_R1W2; K = 32; srcm = pp.r1_W2; }
    else if (i < OFF_R2W2) { base = OFF_R2W1; K = 32; srcm = pp.r2_W1; }
    else                   { base = OFF_R2W2; K = 32; srcm = pp.r2_W2; }
    const int loc = i - base;
    const int n = loc / K, k = loc - n * K;
    const float* g = srcm + k * HD2 + n;
#if ASYNC_LDS
    async_copy_f32(g, &sW[i]);
#else
    sW[i] = *g;
#endif
  }
#if ASYNC_LDS
  __builtin_amdgcn_s_wait_asynccnt(0);
#endif

  { // z1pre = relu(hA[i] + hB[j]); lane -> (row, col-half)
    const int r = lane >> 1;
    const int c0 = (lane & 1) * 32;
    const int p = pairbase + r;
    const int b = p / (NN * NN);
    const int rem = p - b * NN * NN;
    const int i = rem / NN;
    const int j = rem - i * NN;
    const float* pa = hA + (b * NN + i) * HH + c0;
    const float* pb = hB + (b * NN + j) * HH + c0;
    for (int c = 0; c < 32; ++c) sZ[wave][r][c0 + c] = fmaxf(pa[c] + pb[c], 0.f);
  }
  __syncthreads();
  if (lane < 16) { // LN64 with lp_g / lp_t
    float* row = sZ[wave][lane];
    float s = 0.f;
    for (int c = 0; c < 64; ++c) s += row[c];
    const float mu = s * (1.f / 64.f);
    float v = 0.f;
    for (int c = 0; c < 64; ++c) { float d = row[c] - mu; v += d * d; }
    const float inv = rsqrtf(v * (1.f / 64.f) + EPSLN);
    for (int c = 0; c < 64; ++c) row[c] = (row[c] - mu) * inv * pp.lp_g[c] + pp.lp_t[c];
  }
  __syncthreads();

  v8f s0, s1, t0, t1;
  // resblock 1 (projection skip) -- shared A-frag stream for W1 and Ws
  gemm_tiles_dual(&sZ[wave][0][0], 64, sW + OFF_R1W1, pp.r1_b1,
                  sW + OFF_R1WS, pp.r1_bs, 64, lane, t0, t1, s0, s1);
  store_tiles(&sT[wave][0][0], lane, t0, t1);
  __syncthreads();
  ln32(&sT[wave][0][0], pp.r1_g1, pp.r1_t1, lane, true);
  __syncthreads();
  gemm_tiles(&sT[wave][0][0], 32, sW + OFF_R1W2, pp.r1_b2, 32, lane, t0, t1);
  store_tiles(&sU[wave][0][0], lane, t0, t1);
  __syncthreads();
  ln32(&sU[wave][0][0], pp.r1_g2, pp.r1_t2, lane, false);
  __syncthreads();
  add_frag_relu(&sU[wave][0][0], lane, s0, s1); // z2 = relu(o2 + skip)
  __syncthreads();

  // resblock 2 (identity skip = sU)
  gemm_tiles(&sU[wave][0][0], 32, sW + OFF_R2W1, pp.r2_b1, 32, lane, t0, t1);
  store_tiles(&sT[wave][0][0], lane, t0, t1);
  __syncthreads();
  ln32(&sT[wave][0][0], pp.r2_g1, pp.r2_t1, lane, true);
  __syncthreads();
  gemm_tiles(&sT[wave][0][0], 32, sW + OFF_R2W2, pp.r2_b2, 32, lane, t0, t1);
  store_tiles(&sV[wave][0][0], lane, t0, t1);
  __syncthreads();
  ln32(&sV[wave][0][0], pp.r2_g2, pp.r2_t2, lane, false);
  __syncthreads();
  for (int tt = 0; tt < 16; ++tt) { // z3 = relu(o2' + z2)
    const int e = tt * 32 + lane;
    const int r = e >> 5, c = e & 31;
    sV[wave][r][c] = fmaxf(sV[wave][r][c] + sU[wave][r][c], 0.f);
  }
  __syncthreads();
  if (lane < 16) { // out = sigmoid(z3 @ out_W + out_b)
    float acc = pp.out_b[0];
    for (int c = 0; c < 32; ++c) acc += sV[wave][lane][c] * pp.out_W[c];
    out[pairbase + lane] = sigmoidf_(acc);
  }
}

// ================================================================ launch
extern "C" void kernel_launch(void* const* d_in, const int* in_sizes, int n_in,
                              void* d_out, int out_size, void* d_ws, size_t ws_size,
                              hipStream_t stream) {
  (void)in_sizes; (void)n_in; (void)out_size; (void)ws_size;
  const float* x    = (const float*)d_in[0];
  const float* ew   = (const float*)d_in[1];
  const float* in_W = (const float*)d_in[2];
  const float* in_b = (const float*)d_in[3];
  const float* in_g = (const float*)d_in[4];
  const float* in_t = (const float*)d_in[5];
  const float* gcn_W = (const float*)d_in[6];
  const float* gcn_b = (const float*)d_in[7];
  const float* lin_W = (const float*)d_in[8];
  const float* lin_b = (const float*)d_in[9];
  const float* Tx   = (const float*)d_in[10];
  const float* bx   = (const float*)d_in[11];
  /* Th = d_in[12] unused: cheb(0,Th,bh)==bh */
  const float* bh   = (const float*)d_in[13];
  const float* wc   = (const float*)d_in[14];
  const float* bg   = (const float*)d_in[15];
  const float* ln_g = (const float*)d_in[16];
  const float* ln_t = (const float*)d_in[17];
  const float* np_W = (const float*)d_in[18];
  const float* np_b = (const float*)d_in[19];
  const float* np_g = (const float*)d_in[20];
  const float* np_t = (const float*)d_in[21];
  const float* lp_W = (const float*)d_in[22];
  const float* lp_b = (const float*)d_in[23];
  PairParams pp;
  pp.lp_g = (const float*)d_in[24]; pp.lp_t = (const float*)d_in[25];
  pp.r1_W1 = (const float*)d_in[26]; pp.r1_b1 = (const float*)d_in[27];
  pp.r1_W2 = (const float*)d_in[28]; pp.r1_b2 = (const float*)d_in[29];
  pp.r1_g1 = (const float*)d_in[30]; pp.r1_t1 = (const float*)d_in[31];
  pp.r1_g2 = (const float*)d_in[32]; pp.r1_t2 = (const float*)d_in[33];
  pp.r1_Ws = (const float*)d_in[34]; pp.r1_bs = (const float*)d_in[35];
  pp.r2_W1 = (const float*)d_in[36]; pp.r2_b1 = (const float*)d_in[37];
  pp.r2_W2 = (const float*)d_in[38]; pp.r2_b2 = (const float*)d_in[39];
  pp.r2_g1 = (const float*)d_in[40]; pp.r2_t1 = (const float*)d_in[41];
  pp.r2_g2 = (const float*)d_in[42]; pp.r2_t2 = (const float*)d_in[43];
  pp.out_W = (const float*)d_in[44]; pp.out_b = (const float*)d_in[45];
  const int* edge_index = (const int*)d_in[46];
  const int* src = edge_index;
  const int* dst = edge_index + EE;

  // workspace layout
  float* w = (float*)d_ws;
  float* deg   = w;             w += NN;
  float* dinv  = w;             w += NN;
  float* dgi   = w;             w += NN;
  float* gself = w;             w += NN;
  float* cw    = w;             w += EE;
  float* gw    = w;             w += EE;
  float* xp    = w;             w += BN * TT * FF;
  float* h     = w;             w += BNH;
  float* vp    = w;             w += BNH;
  float* Pp    = w;             w += BNH;
  float* bufA  = w;             w += BNH;
  float* bufB  = w;             w += BNH;
  float* bufC  = w;             w += BNH;
  float* P1    = w;             w += BNH;
  float* P2    = w;             w += BNH;
  float* hN    = w;             w += BNH;
  float* hA    = w;             w += BNH;
  float* hB    = w;             w += BNH;

  const int TB = 256;
  // graph normalization
  k_zero<<<(NN + TB - 1) / TB, TB, 0, stream>>>(deg, NN);
  k_deg<<<EE / TB, TB, 0, stream>>>(ew, dst, deg);
  k_nodenorm<<<(NN + TB - 1) / TB, TB, 0, stream>>>(deg, dinv, dgi, gself);
  k_edgew<<<EE / TB, TB, 0, stream>>>(ew, src, dst, dinv, dgi, cw, gw);

  // input projection + LN
  k_pack<<<(BN * TT * FF + TB - 1) / TB, TB, 0, stream>>>(x, xp);
  k_gemm<<<BN / 16, 128, 0, stream>>>(xp, in_W, in_b, h, TT * FF, 0);
  k_row_ln<<<BN, 64, 0, stream>>>(h, in_g, in_t, 1);

  // GRU step (H0=0): Z gate (gcn 0, lin 0) and candidate (gcn 2, lin 2); R unused
  const int gprop = (EE * BB * HH) / TB;
  k_gemm<<<BN / 16, 128, 0, stream>>>(h, gcn_W, nullptr, vp, HH, 0);
  k_zero<<<BNH / TB, TB, 0, stream>>>(Pp, BNH);
  k_prop<<<gprop, TB, 0, stream>>>(Pp, vp, gw, src, dst, 1.f);
  k_gcnfin<<<BNH / TB, TB, 0, stream>>>(Pp, vp, gself, gcn_b);
  k_gemm<<<BN / 16, 128, 0, stream>>>(Pp, lin_W, lin_b, bufA, HH, 0);

  k_gemm<<<BN / 16, 128, 0, stream>>>(h, gcn_W + 2 * HH * HH, nullptr, vp, HH, 0);
  k_zero<<<BNH / TB, TB, 0, stream>>>(Pp, BNH);
  k_prop<<<gprop, TB, 0, stream>>>(Pp, vp, gw, src, dst, 1.f);
  k_gcnfin<<<BNH / TB, TB, 0, stream>>>(Pp, vp, gself, gcn_b + 2 * HH);
  k_gemm<<<BN / 16, 128, 0, stream>>>(Pp, lin_W + 2 * 128 * HH, lin_b + 2 * HH, bufB, HH, 0);

  k_gru<<<BNH / TB, TB, 0, stream>>>(h, bufA, bufB);

  // GConvLSTM stack (zero hidden/cell): gates 0 (I), 2 (C), 3 (O)
  for (int l = 0; l < 2; ++l) {
    k_zero<<<BNH / TB, TB, 0, stream>>>(P1, BNH);
    k_prop<<<gprop, TB, 0, stream>>>(P1, h, cw, src, dst, 1.f);
    k_negcopy<<<BNH / TB, TB, 0, stream>>>(P2, h, BNH);
    k_prop<<<gprop, TB, 0, stream>>>(P2, P1, cw, src, dst, 2.f);

    const int gates[3] = {0, 2, 3};
    float* tgt[3] = {bufA, bufB, bufC};
    for (int gi = 0; gi < 3; ++gi) {
      const int g = gates[gi];
      const float* Txg = Tx + (size_t)((l * 4 + g) * 3) * HH * HH;
      k_gemm<<<BN / 16, 128, 0, stream>>>(h,  Txg,              bx + (l * 4 + g) * HH, tgt[gi], HH, 0);
      k_gemm<<<BN / 16, 128, 0, stream>>>(P1, Txg + HH * HH,     nullptr,               tgt[gi], HH, 1);
      k_gemm<<<BN / 16, 128, 0, stream>>>(P2, Txg + 2 * HH * HH, nullptr,               tgt[gi], HH, 1);
    }
    k_lstm<<<BN, 64, 0, stream>>>(h, bufA, bufB, bufC,
                                  bh + (l * 4 + 0) * HH, bh + (l * 4 + 2) * HH, bh + (l * 4 + 3) * HH,
                                  bg + (l * 4 + 0) * HH, bg + (l * 4 + 2) * HH, bg + (l * 4 + 3) * HH,
                                  wc + (l * 3 + 2) * HH, ln_g + l * HH, ln_t + l * HH);
  }

  // node projection + LN
  k_gemm<<<BN / 16, 128, 0, stream>>>(h, np_W, np_b, hN, HH, 0);
  k_row_ln<<<BN, 64, 0, stream>>>(hN, np_g, np_t, 1);

  // split the concat-GEMM: hA = hN@lp_W[:64], hB = hN@lp_W[64:] + lp_b
  k_gemm<<<BN / 16, 128, 0, stream>>>(hN, lp_W, nullptr, hA, HH, 0);
  k_gemm<<<BN / 16, 128, 0, stream>>>(hN, lp_W + 64 * HH, lp_b, hB, HH, 0);

  // fused all-pairs link predictor
  k_pairs<<<NPAIR / (16 * PW), 32 * PW, 0, stream>>>(hA, hB, pp, (float*)d_out);
}